// FFTDetrend_89043261980969
// MI455X (gfx1250) — compile-verified
//
#include <hip/hip_runtime.h>

#define SEQ   512
#define PRED  96
#define NC    9            // 1 DC + 4 harmonics * (cos, sin)
#define NROWS (64 * 1024)

typedef __attribute__((ext_vector_type(2))) float v2f;
typedef __attribute__((ext_vector_type(8))) float v8f;

// ---------------------------------------------------------------------------
// Kernel 1: build basis tables + folded forecast matrix. One block, 512 thr.
//   analCM : packed analysis basis for WMMA B-operand.
//            analCM[(s>>1)*16 + n][2] = { A[2*(s>>1)][n], A[2*(s>>1)+1][n] }
//   synthT : synthT[t*NC + j] = synthesis basis (includes 1/512 and 2/512)
//   fmat   : fmat[p*NC + j]   = sum_t W[p,t] * synthT[t,j]
// ---------------------------------------------------------------------------
__global__ __launch_bounds__(512) void prep_kernel(const float* __restrict__ W,
                                                   float* __restrict__ analCM,
                                                   float* __restrict__ synthT,
                                                   float* __restrict__ fmat) {
  __shared__ float ssy[SEQ * NC];          // 18 KiB
  const int   t      = threadIdx.x;        // 0..511  (== s, == t)
  const float TWO_PI = 6.28318530717958647692f;
  const float INV_N  = 1.0f / (float)SEQ;

  // synthesis row t
  float sy[NC];
  sy[0] = INV_N;
#pragma unroll
  for (int k = 1; k <= 4; ++k) {
    int   m   = (k * t) & (SEQ - 1);       // exact periodic reduction
    float ang = TWO_PI * ((float)m * INV_N);
    float s, c;
    sincosf(ang, &s, &c);
    sy[2 * k - 1] = (2.0f * INV_N) * c;
    sy[2 * k]     = (2.0f * INV_N) * s;
  }
#pragma unroll
  for (int j = 0; j < NC; ++j) {
    synthT[t * NC + j] = sy[j];
    ssy[t * NC + j]    = sy[j];
  }

  // analysis matrix column n at sample s=t, packed in WMMA-B pair layout
  for (int n = 0; n < 16; ++n) {
    float v = 0.0f;
    if (n == 0) {
      v = 1.0f;
    } else if (n <= 8) {
      int   k   = (n + 1) >> 1;            // 1..4
      int   m   = (k * t) & (SEQ - 1);
      float ang = TWO_PI * ((float)m * INV_N);
      float s, c;
      sincosf(ang, &s, &c);
      v = (n & 1) ? c : s;                 // odd n -> cos, even n -> sin
    }
    analCM[((t >> 1) * 16 + n) * 2 + (t & 1)] = v;
  }
  __syncthreads();

  // fold W through the synthesis basis: fmat[p][j]
  for (int job = t; job < PRED * NC; job += 512) {
    int          p   = job / NC;
    int          j   = job % NC;
    const float* wr  = W + (size_t)p * SEQ;
    float        acc = 0.0f;
    for (int s = 0; s < SEQ; ++s) acc = fmaf(wr[s], ssy[s * NC + j], acc);
    fmat[p * NC + j] = acc;
  }
}

// ---------------------------------------------------------------------------
// Kernel 2: analysis GEMM via V_WMMA_F32_16X16X4_F32.
// Each wave owns a 16-row tile: coeff(16x16) = X(16x512) * A(512x16),
// marching K in steps of 4 (128 WMMAs, 2 accumulators).
// A-operand (16x4 f32): lanes 0-15 K={0,1}, lanes 16-31 K={2,3} per VGPR pair.
// ---------------------------------------------------------------------------
__global__ __launch_bounds__(256) void anal_kernel(const float* __restrict__ x,
                                                   const float* __restrict__ analCM,
                                                   float* __restrict__ coeff) {
  const int lane = threadIdx.x & 31;
  const int wid  = threadIdx.x >> 5;
  const int half = lane >> 4;              // 0: K0/K1, 1: K2/K3
  const int lrow = lane & 15;
  const int tile = blockIdx.x * 8 + wid;   // 16-row tile index
  const int rowBase = tile * 16;

  const float* xrow = x + (size_t)(rowBase + lrow) * SEQ + 2 * half;
  const float* brow = analCM + (half * 16 + lrow) * 2;

  v8f c0 = {};
  v8f c1 = {};
#pragma unroll 4
  for (int kk = 0; kk < 128; kk += 2) {
    v2f a0 = *(const v2f*)(xrow + 4 * kk);
    v2f b0 = *(const v2f*)(brow + 64 * kk);
    c0 = __builtin_amdgcn_wmma_f32_16x16x4_f32(false, a0, false, b0,
                                               (short)0, c0, false, false);
    v2f a1 = *(const v2f*)(xrow + 4 * kk + 4);
    v2f b1 = *(const v2f*)(brow + 64 * kk + 64);
    c1 = __builtin_amdgcn_wmma_f32_16x16x4_f32(false, a1, false, b1,
                                               (short)0, c1, false, false);
  }
  v8f c = c0 + c1;

  // D layout: VGPR r -> row (r + 8*half), col lrow
  float* out = coeff + (size_t)(rowBase + 8 * half) * 16 + lrow;
#pragma unroll
  for (int r = 0; r < 8; ++r) out[(size_t)r * 16] = c[r];
}

// ---------------------------------------------------------------------------
// Kernel 3: one block per row. resid = x - synth*coeff (coalesced), plus the
// 96 folded forecast outputs.
// ---------------------------------------------------------------------------
__global__ __launch_bounds__(512) void post_kernel(const float* __restrict__ x,
                                                   const float* __restrict__ coeff,
                                                   const float* __restrict__ synthT,
                                                   const float* __restrict__ fmat,
                                                   const float* __restrict__ bias,
                                                   float* __restrict__ resid,
                                                   float* __restrict__ fc) {
  __shared__ float cf[NC];
  const int row = blockIdx.x;
  const int t   = threadIdx.x;             // 0..511
  if (t < NC) cf[t] = coeff[(size_t)row * 16 + t];
  __syncthreads();

  const float* sy    = synthT + t * NC;
  float        trend = cf[0] * sy[0];
#pragma unroll
  for (int j = 1; j < NC; ++j) trend = fmaf(cf[j], sy[j], trend);

  const size_t idx = (size_t)row * SEQ + t;
  resid[idx] = x[idx] - trend;

  if (t < PRED) {
    const float* fm  = fmat + t * NC;
    float        acc = bias[t];
#pragma unroll
    for (int j = 0; j < NC; ++j) acc = fmaf(cf[j], fm[j], acc);
    fc[(size_t)row * PRED + t] = acc;
  }
}

// ---------------------------------------------------------------------------
extern "C" void kernel_launch(void* const* d_in, const int* in_sizes, int n_in,
                              void* d_out, int out_size, void* d_ws, size_t ws_size,
                              hipStream_t stream) {
  (void)in_sizes; (void)n_in; (void)out_size; (void)ws_size;
  const float* x = (const float*)d_in[0];
  const float* W = (const float*)d_in[1];
  const float* b = (const float*)d_in[2];

  float* fc    = (float*)d_out;                       // (64*1024, 96)
  float* resid = fc + (size_t)NROWS * PRED;           // (64*1024, 512)

  float* ws     = (float*)d_ws;
  float* analCM = ws;                                 // 256*16*2   = 8192 f
  float* synthT = ws + 8192;                          // 512*9      = 4608 f
  float* fmat   = ws + 8192 + 4608;                   // 96*9       = 864  f
  float* coeff  = ws + 16384;                         // 65536*16   = 1M   f

  prep_kernel<<<1, 512, 0, stream>>>(W, analCM, synthT, fmat);
  anal_kernel<<<NROWS / 128, 256, 0, stream>>>(x, analCM, coeff);
  post_kernel<<<NROWS, 512, 0, stream>>>(x, coeff, synthT, fmat, b, resid, fc);
}